// Qwen3_5Attention_33028298506809
// MI455X (gfx1250) — compile-verified
//
#include <hip/hip_runtime.h>
#include <math.h>

#define NUM_HEADS 32
#define NUM_KV_HEADS 8
#define HEAD_DIM 128
#define HIDDEN 4096
#define BSZ 32
#define MAX_SEQ 4096
#define QKV_OUT ((NUM_HEADS + 2 * NUM_KV_HEADS) * HEAD_DIM) /* 6144 */
#define GROUP (NUM_HEADS / NUM_KV_HEADS)                    /* 4 */
#define ATT_SCALE 0.08838834764831845f                      /* 128^-0.5 */
#define KSPLIT 4

typedef __attribute__((ext_vector_type(2))) float v2f;
typedef __attribute__((ext_vector_type(4))) float v4f;
typedef __attribute__((ext_vector_type(8))) float v8f;

// ---------------------------------------------------------------------------
// Y[kc][32, N] = X[32, kstart:kstart+Kc] @ W[N, kstart:kstart+Kc]^T (+bias).
// blockIdx.y = K-chunk index (kc); Y is offset by kc*32*N so each chunk has a
// private partial buffer (direct path uses gridDim.y == 1 and bias != null).
// fp32 WMMA 16x16x4; one wave owns a 16-wide N tile and both M=16 tiles so
// every W byte is loaded exactly once per chunk.
//
// Software-pipelined: the K loop runs in groups of 4 K-steps (16 floats =
// one 64B cacheline per row). The next group's 12 loads are issued into a
// separate register set BEFORE the current group's 8 WMMAs consume their
// fragments, keeping ~12 loads outstanding per wave at all times.
//
// A frag (16x4): lane&15 = M row, lane>>4 selects K pair {0,1}|{2,3} -> v2f.
// B frag (4x16): lane&15 = N col, same K-pair split -> v2f (contiguous load).
// C/D (16x16):   vgpr r -> M = r + 8*(lane>>4), N = lane&15.
// ---------------------------------------------------------------------------
__global__ __launch_bounds__(256) void gemm_wmma_f32(
    const float* __restrict__ X, const float* __restrict__ W,
    const float* __restrict__ bias, float* __restrict__ Y, int N, int ldk,
    int Kc) {
  const int lane = threadIdx.x & 31;
  const int wave = threadIdx.x >> 5;
  const int n0 = (blockIdx.x * 8 + wave) * 16;
  if (n0 >= N) return;

  const int kstart = blockIdx.y * Kc;
  float* __restrict__ Yb = Y + (size_t)blockIdx.y * 32 * N;

  const int mlo = lane & 15;          // M (A) / N (B) within tile
  const int khalf = (lane >> 4) * 2;  // K pair selector

  const float bini = bias ? bias[n0 + mlo] : 0.0f;  // per-column bias
  v8f acc0, acc1;
#pragma unroll
  for (int r = 0; r < 8; ++r) { acc0[r] = bini; acc1[r] = bini; }

  const float* xrow0 = X + (size_t)mlo * ldk + kstart + khalf;
  const float* xrow1 = X + (size_t)(mlo + 16) * ldk + kstart + khalf;
  const float* wrow = W + (size_t)(n0 + mlo) * ldk + kstart + khalf;

  // ---- pipelined main loop: groups of 4 K-steps (16 K) ----
  v2f a0s[4], a1s[4], bbs[4];
#pragma unroll
  for (int i = 0; i < 4; ++i) {  // prologue: group 0
    a0s[i] = *(const v2f*)(xrow0 + i * 4);
    a1s[i] = *(const v2f*)(xrow1 + i * 4);
    bbs[i] = *(const v2f*)(wrow + i * 4);
  }

  for (int kb = 0; kb + 16 < Kc; kb += 16) {
    v2f na0[4], na1[4], nbb[4];
#pragma unroll
    for (int i = 0; i < 4; ++i) {  // issue next group's 12 loads first
      const int kk = kb + 16 + i * 4;
      na0[i] = *(const v2f*)(xrow0 + kk);
      na1[i] = *(const v2f*)(xrow1 + kk);
      nbb[i] = *(const v2f*)(wrow + kk);
    }
#pragma unroll
    for (int i = 0; i < 4; ++i) {  // consume current group
      acc0 = __builtin_amdgcn_wmma_f32_16x16x4_f32(
          false, a0s[i], false, bbs[i], (short)0, acc0, false, false);
      acc1 = __builtin_amdgcn_wmma_f32_16x16x4_f32(
          false, a1s[i], false, bbs[i], (short)0, acc1, false, false);
    }
#pragma unroll
    for (int i = 0; i < 4; ++i) {  // rotate buffers
      a0s[i] = na0[i];
      a1s[i] = na1[i];
      bbs[i] = nbb[i];
    }
  }
#pragma unroll
  for (int i = 0; i < 4; ++i) {  // epilogue: last group
    acc0 = __builtin_amdgcn_wmma_f32_16x16x4_f32(false, a0s[i], false, bbs[i],
                                                 (short)0, acc0, false, false);
    acc1 = __builtin_amdgcn_wmma_f32_16x16x4_f32(false, a1s[i], false, bbs[i],
                                                 (short)0, acc1, false, false);
  }

  const int mtop = (lane >> 4) * 8;
#pragma unroll
  for (int r = 0; r < 8; ++r) {
    const int m = mtop + r;
    Yb[(size_t)m * N + n0 + mlo] = acc0[r];
    Yb[(size_t)(m + 16) * N + n0 + mlo] = acc1[r];
  }
}

// Deterministic fixed-order combine of the K-split partials (+ bias).
__global__ __launch_bounds__(256) void reduce_ksplit(
    const float* __restrict__ part, const float* __restrict__ bias,
    float* __restrict__ Y, int N, int S) {
  const int idx = blockIdx.x * 256 + threadIdx.x;
  if (idx >= 32 * N) return;
  const int n = idx % N;
  float acc = bias ? bias[n] : 0.0f;
  for (int s = 0; s < S; ++s) acc += part[(size_t)s * 32 * N + idx];
  Y[idx] = acc;
}

// ---------------------------------------------------------------------------
// Flash-decode attention. Block = (batch, kv_head), 8 waves split the
// sequence; per-lane float4 of the 128-dim head (b128 loads), online softmax
// per grouped query head, deterministic LDS merge (no float atomics).
// Main loop processes 4 positions per trip (8 b128 loads in flight); the
// scalar tail handles the remainder including the freshly projected K/V at
// position seq_len-1 (input cache is read-only).
// ---------------------------------------------------------------------------
__global__ __launch_bounds__(256) void attn_decode(
    const float* __restrict__ qkv,       // [32, 6144]
    const float* __restrict__ kv_cache,  // [2, 32, 4096, 8, 128]
    const int* __restrict__ seq_lens,
    float* __restrict__ attn_out) {      // [32, 4096]
  const int b = blockIdx.x / NUM_KV_HEADS;
  const int kh = blockIdx.x % NUM_KV_HEADS;
  const int lane = threadIdx.x & 31;
  const int wave = threadIdx.x >> 5;
  const int L = seq_lens[b];

  const float* qb = qkv + (size_t)b * QKV_OUT;
  v4f qv[GROUP];
#pragma unroll
  for (int g = 0; g < GROUP; ++g)
    qv[g] = *(const v4f*)(qb + (kh * GROUP + g) * HEAD_DIM + lane * 4);

  const float* knew = qb + NUM_HEADS * HEAD_DIM + kh * HEAD_DIM;
  const float* vnew = qb + (NUM_HEADS + NUM_KV_HEADS) * HEAD_DIM + kh * HEAD_DIM;

  const size_t srow = (size_t)NUM_KV_HEADS * HEAD_DIM;  // stride per position
  const float* kcache =
      kv_cache + ((size_t)b * MAX_SEQ * NUM_KV_HEADS + kh) * HEAD_DIM;
  const float* vcache =
      kcache + (size_t)BSZ * MAX_SEQ * NUM_KV_HEADS * HEAD_DIM;

  float m[GROUP], l[GROUP];
  v4f o[GROUP];
#pragma unroll
  for (int g = 0; g < GROUP; ++g) {
    m[g] = -INFINITY;
    l[g] = 0.0f;
    o[g] = (v4f)(0.0f);
  }

  auto process = [&](const v4f kvec, const v4f vvec) {
#pragma unroll
    for (int g = 0; g < GROUP; ++g) {
      float d = kvec.x * qv[g].x + kvec.y * qv[g].y + kvec.z * qv[g].z +
                kvec.w * qv[g].w;
#pragma unroll
      for (int off = 16; off > 0; off >>= 1) d += __shfl_xor(d, off, 32);
      const float sc = d * ATT_SCALE;
      const float mn = fmaxf(m[g], sc);
      const float c = __expf(m[g] - mn);
      const float e = __expf(sc - mn);
      l[g] = l[g] * c + e;
      o[g] = o[g] * c + e * vvec;
      m[g] = mn;
    }
  };

  int s = wave;
  // Main loop: 4 positions per trip, all strictly below L-1 (pure cache).
  for (; s + 24 < L - 1; s += 32) {
    const float* kp = kcache + (size_t)s * srow + lane * 4;
    const float* vp = vcache + (size_t)s * srow + lane * 4;
    const v4f k0 = *(const v4f*)(kp);
    const v4f k1 = *(const v4f*)(kp + 8 * srow);
    const v4f k2 = *(const v4f*)(kp + 16 * srow);
    const v4f k3 = *(const v4f*)(kp + 24 * srow);
    const v4f w0 = *(const v4f*)(vp);
    const v4f w1 = *(const v4f*)(vp + 8 * srow);
    const v4f w2 = *(const v4f*)(vp + 16 * srow);
    const v4f w3 = *(const v4f*)(vp + 24 * srow);
    process(k0, w0);
    process(k1, w1);
    process(k2, w2);
    process(k3, w3);
  }
  // Tail: remaining positions, including the fresh K/V at s == L-1.
  for (; s < L; s += 8) {
    const bool fresh = (s == L - 1);
    const v4f kvec = fresh ? *(const v4f*)(knew + lane * 4)
                           : *(const v4f*)(kcache + (size_t)s * srow + lane * 4);
    const v4f vvec = fresh ? *(const v4f*)(vnew + lane * 4)
                           : *(const v4f*)(vcache + (size_t)s * srow + lane * 4);
    process(kvec, vvec);
  }

  __shared__ float s_m[8][GROUP];
  __shared__ float s_l[8][GROUP];
  __shared__ float s_o[8][GROUP][HEAD_DIM];  // 16 KB

  if (lane == 0) {
#pragma unroll
    for (int g = 0; g < GROUP; ++g) {
      s_m[wave][g] = m[g];
      s_l[wave][g] = l[g];
    }
  }
#pragma unroll
  for (int g = 0; g < GROUP; ++g)
    *(v4f*)&s_o[wave][g][lane * 4] = o[g];
  __syncthreads();

  for (int flat = threadIdx.x; flat < GROUP * HEAD_DIM; flat += 256) {
    const int g = flat >> 7;
    const int d = flat & (HEAD_DIM - 1);
    float M = -INFINITY;
#pragma unroll
    for (int w = 0; w < 8; ++w) M = fmaxf(M, s_m[w][g]);
    float lsum = 0.0f, osum = 0.0f;
#pragma unroll
    for (int w = 0; w < 8; ++w) {  // fixed order -> deterministic
      const float f = __expf(s_m[w][g] - M);
      lsum += s_l[w][g] * f;
      osum += s_o[w][g][d] * f;
    }
    attn_out[(size_t)b * (NUM_HEADS * HEAD_DIM) + (kh * GROUP + g) * HEAD_DIM +
             d] = osum / lsum;
  }
}

// ---------------------------------------------------------------------------
extern "C" void kernel_launch(void* const* d_in, const int* in_sizes, int n_in,
                              void* d_out, int out_size, void* d_ws,
                              size_t ws_size, hipStream_t stream) {
  const float* hidden = (const float*)d_in[0];  // [32,1,4096]
  const float* kv_cache = (const float*)d_in[2];
  const int* seq_lens = (const int*)d_in[4];
  const float* Wqkv = (const float*)d_in[5];  // [6144,4096]
  const float* bqkv = (const float*)d_in[6];
  const float* Wo = (const float*)d_in[7];    // [4096,4096]
  float* out = (float*)d_out;

  const size_t qkv_elems = (size_t)BSZ * QKV_OUT;    // 196,608
  const size_t attn_elems = (size_t)BSZ * HIDDEN;    // 131,072
  const size_t part_elems = (size_t)KSPLIT * BSZ * QKV_OUT;  // max-N partials
  const size_t need = (qkv_elems + attn_elems + part_elems) * sizeof(float);

  float* qkv = (float*)d_ws;
  float* attn_out = qkv + qkv_elems;
  float* part = attn_out + attn_elems;
  const bool split = (ws_size >= need);  // constant per run -> deterministic

  // 1) QKV projection: qkv = x @ Wqkv^T + bqkv
  if (split) {
    gemm_wmma_f32<<<dim3(QKV_OUT / 128, KSPLIT), 256, 0, stream>>>(
        hidden, Wqkv, nullptr, part, QKV_OUT, HIDDEN, HIDDEN / KSPLIT);
    reduce_ksplit<<<(32 * QKV_OUT + 255) / 256, 256, 0, stream>>>(
        part, bqkv, qkv, QKV_OUT, KSPLIT);
  } else {
    gemm_wmma_f32<<<dim3(QKV_OUT / 128, 1), 256, 0, stream>>>(
        hidden, Wqkv, bqkv, qkv, QKV_OUT, HIDDEN, HIDDEN);
  }

  // 2) attention over KV cache
  attn_decode<<<BSZ * NUM_KV_HEADS, 256, 0, stream>>>(qkv, kv_cache, seq_lens,
                                                      attn_out);

  // 3) output projection: out = attn @ Wo^T
  if (split) {
    gemm_wmma_f32<<<dim3(HIDDEN / 128, KSPLIT), 256, 0, stream>>>(
        attn_out, Wo, nullptr, part, HIDDEN, HIDDEN, HIDDEN / KSPLIT);
    reduce_ksplit<<<(32 * HIDDEN + 255) / 256, 256, 0, stream>>>(
        part, nullptr, out, HIDDEN, KSPLIT);
  } else {
    gemm_wmma_f32<<<dim3(HIDDEN / 128, 1), 256, 0, stream>>>(
        attn_out, Wo, nullptr, out, HIDDEN, HIDDEN, HIDDEN);
  }
}